// TaylorPolynomial_11424613007381
// MI455X (gfx1250) — compile-verified
//
#include <hip/hip_runtime.h>
#include <hip/hip_bf16.h>

// Vector types for the WMMA operands.
typedef __attribute__((ext_vector_type(2))) float v2f;   // A (16x4 f32) / B (4x16 f32): 2 VGPRs
typedef __attribute__((ext_vector_type(4))) float v4f;   // b128 load payload
typedef __attribute__((ext_vector_type(8))) float v8f;   // C/D (16x16 f32): 8 VGPRs

#define WAVES_PER_BLOCK 8
#define THREADS_PER_BLOCK (WAVES_PER_BLOCK * 32)

// ---------------------------------------------------------------------------
// Kernel 1: per-term scale[t] = prod_j delta[j]^e_j / prod_j e_j!
// Integer-exponent semantics via repeated multiply (0^0 == 1, negative bases
// exact), exact 1/e! table (e <= DEGREE = 8 guaranteed since sum(e) <= 8).
// ---------------------------------------------------------------------------
__global__ void taylor_scale_kernel(const float* __restrict__ x,
                                    const float* __restrict__ ep,
                                    const int*   __restrict__ orders,
                                    float*       __restrict__ scales,
                                    int T) {
    int t = blockIdx.x * blockDim.x + threadIdx.x;
    if (t >= T) return;
    const float inv_fact[9] = {1.0f, 1.0f, 1.0f / 2.0f, 1.0f / 6.0f,
                               1.0f / 24.0f, 1.0f / 120.0f, 1.0f / 720.0f,
                               1.0f / 5040.0f, 1.0f / 40320.0f};
    float p = 1.0f;
#pragma unroll
    for (int j = 0; j < 8; ++j) {
        float d = x[j] - ep[j];
        int e = orders[t * 8 + j];
        if (e < 0) e = 0;
        if (e > 8) e = 8;
        float pw = 1.0f;
        for (int k = 0; k < e; ++k) pw *= d;   // exact int-pow semantics
        p *= pw * inv_fact[e];
    }
    scales[t] = p;
}

// ---------------------------------------------------------------------------
// Kernel 2: streaming weighted row-sum via V_WMMA_F32_16X16X4_F32.
// One wave per term-row (4096 contiguous floats = 16 KB). A holds scale[t] in
// every slot, so sum(D) == 16 * scale[t] * sum(row) regardless of B layout.
// One block partial per block (fixed-order reduction, no atomics).
// ---------------------------------------------------------------------------
__global__ void taylor_reduce_kernel(const float* __restrict__ deriv,
                                     const float* __restrict__ scales,
                                     float*       __restrict__ partials,
                                     int T, int C) {
    __shared__ float smem[WAVES_PER_BLOCK];
    const int lane = threadIdx.x & 31;
    const int wave = threadIdx.x >> 5;
    const int t = blockIdx.x * WAVES_PER_BLOCK + wave;

    float wave_sum = 0.0f;
    if (t < T) {                       // wave-uniform: EXEC stays all-1s inside
        const float s = scales[t];
        v2f a; a.x = s; a.y = s;       // scale broadcast into all A slots
        v8f acc = {};
        const float* row = deriv + (size_t)t * (size_t)C;
        // 128 floats per iteration: each lane one b128 load (512 B / wave),
        // feeding two 4x16 f32 B-matrices -> two chained WMMAs into acc.
#pragma unroll 4
        for (int i = 0; i < C; i += 128) {
            v4f d = *(const v4f*)(row + i + lane * 4);
            v2f b0; b0.x = d.x; b0.y = d.y;
            v2f b1; b1.x = d.z; b1.y = d.w;
            acc = __builtin_amdgcn_wmma_f32_16x16x4_f32(
                false, a, false, b0, (short)0, acc, false, false);
            acc = __builtin_amdgcn_wmma_f32_16x16x4_f32(
                false, a, false, b1, (short)0, acc, false, false);
        }
        // Sum the 8 accumulator VGPRs per lane, then fixed-order lane tree.
        float lsum = ((acc[0] + acc[1]) + (acc[2] + acc[3])) +
                     ((acc[4] + acc[5]) + (acc[6] + acc[7]));
#pragma unroll
        for (int off = 16; off >= 1; off >>= 1)
            lsum += __shfl_xor(lsum, off, 32);
        wave_sum = lsum;
    }
    if (lane == 0) smem[wave] = wave_sum;   // inactive tail waves write 0.0f
    __syncthreads();
    if (threadIdx.x == 0) {
        float bs = 0.0f;
#pragma unroll
        for (int w = 0; w < WAVES_PER_BLOCK; ++w) bs += smem[w];
        partials[blockIdx.x] = bs;
    }
}

// ---------------------------------------------------------------------------
// Kernel 3: deterministic final reduction of block partials; /16 undoes the
// all-rows A broadcast factor (exact power-of-two scale).
// ---------------------------------------------------------------------------
__global__ void taylor_final_kernel(const float* __restrict__ partials,
                                    int n, float* __restrict__ out) {
    __shared__ float smem[256];
    float s = 0.0f;
    for (int i = threadIdx.x; i < n; i += 256) s += partials[i];
    smem[threadIdx.x] = s;
    __syncthreads();
#pragma unroll
    for (int stride = 128; stride >= 1; stride >>= 1) {
        if ((int)threadIdx.x < stride) smem[threadIdx.x] += smem[threadIdx.x + stride];
        __syncthreads();
    }
    if (threadIdx.x == 0) out[0] = smem[0] * 0.0625f;   // * 1/16
}

// ---------------------------------------------------------------------------
// Launch: inputs in setup_inputs() order: x[8], expansion_point[8],
// derivatives[T*4096], orders[T*8]. Output: single float32 scalar.
// ---------------------------------------------------------------------------
extern "C" void kernel_launch(void* const* d_in, const int* in_sizes, int n_in,
                              void* d_out, int out_size, void* d_ws, size_t ws_size,
                              hipStream_t stream) {
    const float* x      = (const float*)d_in[0];
    const float* ep     = (const float*)d_in[1];
    const float* deriv  = (const float*)d_in[2];
    const int*   orders = (const int*)d_in[3];

    const int T = in_sizes[3] / 8;          // 12870
    const int C = in_sizes[2] / T;          // 4096

    float* scales   = (float*)d_ws;         // T floats
    float* partials = scales + T;           // nblocks2 floats
    float* out      = (float*)d_out;

    const int nblocks1 = (T + 255) / 256;
    const int nblocks2 = (T + WAVES_PER_BLOCK - 1) / WAVES_PER_BLOCK;

    taylor_scale_kernel<<<nblocks1, 256, 0, stream>>>(x, ep, orders, scales, T);
    taylor_reduce_kernel<<<nblocks2, THREADS_PER_BLOCK, 0, stream>>>(
        deriv, scales, partials, T, C);
    taylor_final_kernel<<<1, 256, 0, stream>>>(partials, nblocks2, out);
}